// Attn_loc_freq_71090298683717
// MI455X (gfx1250) — compile-verified
//
#include <hip/hip_runtime.h>
#include <stdint.h>

// Problem geometry (fixed by the reference)
#define N_POI   100000
#define N_BINS  168            // floats per row; 672 bytes = 42 x float4
#define BATCH   1024
#define SEQ     200
#define ROWS    (BATCH * SEQ)  // 204800 output rows

#define WAVES_PER_BLOCK 8
#define THREADS_PER_BLOCK (WAVES_PER_BLOCK * 32)
#define ROWS_PER_WAVE   2
#define ROW_F4          (N_BINS / 4)      // 42 float4 per row
#define TAIL_F4         (ROW_F4 - 32)     // 10 float4 handled by lanes 0..9

typedef float v4f __attribute__((ext_vector_type(4)));
// Exact pointee type the async builtin wants (per hipcc diagnostic):
// "__attribute__((__vector_size__(4 * sizeof(int)))) int" in AS1 / AS3.
typedef int v4i __attribute__((vector_size(16)));
typedef __attribute__((address_space(1))) v4i gv4i;   // global-memory v4i
typedef __attribute__((address_space(3))) v4i lv4i;   // LDS v4i

// ---- CDNA5 async global->LDS copy path (gfx1250 GLOBAL_LOAD_ASYNC_TO_LDS_B128,
// tracked by ASYNCcnt / s_wait_asynccnt). Guarded so the file always compiles.
#if __has_builtin(__builtin_amdgcn_global_load_async_to_lds_b128) && \
    __has_builtin(__builtin_amdgcn_s_wait_asynccnt)
#define USE_ASYNC_LDS 1
#else
#define USE_ASYNC_LDS 0
#endif

#if USE_ASYNC_LDS
__device__ __forceinline__ void async_cp16(const float* g, float* lds) {
  // Per-lane 16B transfer from global memory into LDS.  AS3 pointers are 32-bit,
  // so the integer round-trip keeps exactly the wave-relative LDS offset.
  __builtin_amdgcn_global_load_async_to_lds_b128(
      (gv4i*)(uintptr_t)g,
      (lv4i*)(uint32_t)(uintptr_t)lds,
      /*imm offset*/ 0, /*cpol*/ 0);
}

__device__ __forceinline__ void issue_row_async(const float* g, float* lds, int lane) {
  // 672B row = 32 lanes * 16B  +  10 lanes * 16B (partial EXEC on the tail).
  async_cp16(g + lane * 4, lds + lane * 4);
  if (lane < TAIL_F4)
    async_cp16(g + 128 + lane * 4, lds + 128 + lane * 4);
}
#endif

// Softmax one 168-wide row held at `src` (global or LDS pointer), write
// normalized probabilities to `dst` with non-temporal b128 stores so the
// 137MB write-once output stream does not evict the L2-resident freq table.
__device__ __forceinline__ void softmax_row(const float* __restrict__ src,
                                            float* __restrict__ dst, int lane) {
  const v4f* s4 = (const v4f*)src;
  const bool tail = lane < TAIL_F4;
  const float ninf = -__builtin_inff();

  v4f a = s4[lane];                       // floats 0..127 across the wave
  v4f b;
  if (tail) b = s4[32 + lane];            // floats 128..167 on lanes 0..9
  else      b = (v4f){ninf, ninf, ninf, ninf};

  // wave32 max reduction
  float m = fmaxf(fmaxf(fmaxf(a.x, a.y), fmaxf(a.z, a.w)),
                  fmaxf(fmaxf(b.x, b.y), fmaxf(b.z, b.w)));
#pragma unroll
  for (int s = 16; s > 0; s >>= 1) m = fmaxf(m, __shfl_xor(m, s, 32));

  v4f ea, eb;
  ea.x = __expf(a.x - m); ea.y = __expf(a.y - m);
  ea.z = __expf(a.z - m); ea.w = __expf(a.w - m);
  eb.x = __expf(b.x - m); eb.y = __expf(b.y - m);   // exp(-inf - m) == 0 pads
  eb.z = __expf(b.z - m); eb.w = __expf(b.w - m);

  float t = (ea.x + ea.y) + (ea.z + ea.w) + (eb.x + eb.y) + (eb.z + eb.w);
#pragma unroll
  for (int s = 16; s > 0; s >>= 1) t += __shfl_xor(t, s, 32);

  const float inv = 1.0f / t;
  v4f* d4 = (v4f*)dst;
  v4f oa = {ea.x * inv, ea.y * inv, ea.z * inv, ea.w * inv};
  __builtin_nontemporal_store(oa, d4 + lane);
  if (tail) {
    v4f ob = {eb.x * inv, eb.y * inv, eb.z * inv, eb.w * inv};
    __builtin_nontemporal_store(ob, d4 + 32 + lane);
  }
}

__global__ __launch_bounds__(THREADS_PER_BLOCK)
void attn_softgather_kernel(const int* __restrict__ idx,
                            const float* __restrict__ freq,
                            float* __restrict__ out) {
  const int lane = threadIdx.x & 31;
  const int wv   = threadIdx.x >> 5;
  // Grid is sized exactly: ROWS == gridDim.x * WAVES_PER_BLOCK * ROWS_PER_WAVE
  const long long r0 =
      ((long long)blockIdx.x * WAVES_PER_BLOCK + wv) * ROWS_PER_WAVE;

  const int poi0 = __builtin_amdgcn_readfirstlane(idx[r0]);
  const int poi1 = __builtin_amdgcn_readfirstlane(idx[r0 + 1]);
  const float* s0 = freq + (long long)poi0 * N_BINS;
  const float* s1 = freq + (long long)poi1 * N_BINS;
  float* d0 = out + r0 * N_BINS;
  float* d1 = d0 + N_BINS;

#if USE_ASYNC_LDS
  __shared__ __align__(16) float smem[WAVES_PER_BLOCK][ROWS_PER_WAVE][N_BINS];
  // Double-buffered async staging: both rows' copies in flight, then
  // partial wait (<=2 outstanding) lets row-1 DMA overlap row-0 compute.
  issue_row_async(s0, &smem[wv][0][0], lane);
  issue_row_async(s1, &smem[wv][1][0], lane);
  __builtin_amdgcn_s_wait_asynccnt(2);   // row 0's two copies done (in-order)
  asm volatile("" ::: "memory");
  softmax_row(&smem[wv][0][0], d0, lane);
  __builtin_amdgcn_s_wait_asynccnt(0);   // row 1 done
  asm volatile("" ::: "memory");
  softmax_row(&smem[wv][1][0], d1, lane);
#else
  softmax_row(s0, d0, lane);
  softmax_row(s1, d1, lane);
#endif
}

extern "C" void kernel_launch(void* const* d_in, const int* in_sizes, int n_in,
                              void* d_out, int out_size, void* d_ws, size_t ws_size,
                              hipStream_t stream) {
  (void)in_sizes; (void)n_in; (void)out_size; (void)d_ws; (void)ws_size;
  // d_in[0] = venueid2coor (unused by the reference output)
  const int*   idx  = (const int*)d_in[1];    // inputs_wekn  [1024*200]
  const float* freq = (const float*)d_in[2];  // poi_freq_matrix [100000*168]
  float* out = (float*)d_out;                 // [1024*200*168] f32

  static_assert(ROWS % (WAVES_PER_BLOCK * ROWS_PER_WAVE) == 0, "exact grid");
  const int blocks = ROWS / (WAVES_PER_BLOCK * ROWS_PER_WAVE);  // 12800
  attn_softgather_kernel<<<blocks, THREADS_PER_BLOCK, 0, stream>>>(idx, freq, out);
}